// IndRNN_onlyrecurrent_81956565942496
// MI455X (gfx1250) — compile-verified
//
#include <hip/hip_runtime.h>
#include <stdint.h>

// ---------------------------------------------------------------------------
// IndRNN recurrent-only: h_t = relu(x_t + w ⊙ h_{t-1})
// x: [T=2048, B=64, H=512] f32, w: [H] f32, out: [T, B, H] f32
// Bandwidth-bound (0.25 FLOP/byte): stream x via async global->LDS pipeline
// (ASYNCcnt), write out with non-temporal stores. No matmul -> no WMMA.
// ---------------------------------------------------------------------------

typedef __attribute__((ext_vector_type(2))) float v2f;
typedef __attribute__((ext_vector_type(2))) int   v2i;

#define AS3 __attribute__((address_space(3)))

#define DEPTH 32   // async pipeline stages in flight (ASYNCcnt is 6 bits, <=63)
#define BLOCK 256  // 8 wave32 waves per workgroup

static constexpr int T   = 2048;
static constexpr int B   = 64;
static constexpr int H   = 512;
static constexpr int BH2 = (B * H) / 2;  // float2 elements per timestep = 16384

// ---- CDNA5 async global->LDS copy (8 bytes per lane) ----------------------
__device__ __forceinline__ void async_copy_b64(const v2f* g, v2f* l) {
#if __has_builtin(__builtin_amdgcn_global_load_async_to_lds_b64)
  // Builtin params are int2* (global source, generic) and LDS int2* dest,
  // plus imm offset and imm cpol.
  __builtin_amdgcn_global_load_async_to_lds_b64((v2i*)g, (AS3 v2i*)l, 0, 0);
#else
  unsigned lds_addr = (unsigned)(uintptr_t)(AS3 void*)l;
  asm volatile("global_load_async_to_lds_b64 %0, %1, off"
               :
               : "v"(lds_addr), "v"((unsigned long long)(uintptr_t)g)
               : "memory");
#endif
}

#if __has_builtin(__builtin_amdgcn_s_wait_asynccnt)
#define WAIT_ASYNCCNT(n) __builtin_amdgcn_s_wait_asynccnt(n)
#else
#define WAIT_ASYNCCNT(n) asm volatile("s_wait_asynccnt %0" ::"n"(n) : "memory")
#endif

#if __has_builtin(__builtin_amdgcn_s_wait_dscnt)
#define WAIT_DSCNT(n) __builtin_amdgcn_s_wait_dscnt(n)
#else
#define WAIT_DSCNT(n) asm volatile("s_wait_dscnt %0" ::"n"(n) : "memory")
#endif

__global__ __launch_bounds__(BLOCK) void indrnn_recur_kernel(
    const float* __restrict__ x, const float* __restrict__ w,
    float* __restrict__ out) {
  // One lane owns one float2 channel pair; each lane has a private LDS slot
  // per pipeline stage -> no workgroup barriers needed at all.
  __shared__ v2f buf[DEPTH * BLOCK];  // 64 KB (<< 320 KB/WGP)

  const int tid = threadIdx.x;
  const int col = blockIdx.x * BLOCK + tid;  // [0, BH2)
  const int j   = col & (H / 2 - 1);         // index within hidden dim (v2f units)

  v2f wv;
  wv.x = w[2 * j + 0];
  wv.y = w[2 * j + 1];

  const v2f* __restrict__ xg = (const v2f*)x + col;  // +t*BH2 per step
  v2f* __restrict__ og       = (v2f*)out + col;
  v2f* lbase                 = &buf[tid];  // stage stride = BLOCK

  // ---- prologue: fill DEPTH stages of the async pipeline ----
#pragma unroll
  for (int s = 0; s < DEPTH; ++s)
    async_copy_b64(xg + (size_t)s * BH2, lbase + s * BLOCK);

  v2f h;
  h.x = 0.f;
  h.y = 0.f;
  int slot = 0;

  // ---- steady state: retire oldest stage, refill, recur, NT-store ----
  for (int t = 0; t < T - DEPTH; ++t) {
    WAIT_ASYNCCNT(DEPTH - 1);            // async loads complete in order:
                                         // stage for t has landed in LDS
    v2f xv = buf[slot * BLOCK + tid];    // ds_load_b64 from own slot
    WAIT_DSCNT(0);                       // value in VGPRs -> slot reusable
    async_copy_b64(xg + (size_t)(t + DEPTH) * BH2, lbase + slot * BLOCK);

    v2f z = xv + wv * h;                 // v_fma_f32 x2
    h.x   = fmaxf(z.x, 0.f);
    h.y   = fmaxf(z.y, 0.f);
    __builtin_nontemporal_store(h, og + (size_t)t * BH2);  // written once, never re-read
    slot = (slot + 1) & (DEPTH - 1);
  }

  // ---- epilogue: drain remaining stages ----
  WAIT_ASYNCCNT(0);
  for (int t = T - DEPTH; t < T; ++t) {
    v2f xv = buf[slot * BLOCK + tid];
    v2f z  = xv + wv * h;
    h.x    = fmaxf(z.x, 0.f);
    h.y    = fmaxf(z.y, 0.f);
    __builtin_nontemporal_store(h, og + (size_t)t * BH2);
    slot = (slot + 1) & (DEPTH - 1);
  }
}

extern "C" void kernel_launch(void* const* d_in, const int* in_sizes, int n_in,
                              void* d_out, int out_size, void* d_ws,
                              size_t ws_size, hipStream_t stream) {
  const float* x = (const float*)d_in[0];      // [T, B, H]
  const float* w = (const float*)d_in[1];      // [H]
  float* out     = (float*)d_out;              // [T, B, H]

  dim3 grid(BH2 / BLOCK);  // 64 blocks, exact cover -> EXEC all-ones
  dim3 block(BLOCK);
  indrnn_recur_kernel<<<grid, block, 0, stream>>>(x, w, out);
}